// xgnn_poly_23313082483601
// MI455X (gfx1250) — compile-verified
//
#include <hip/hip_runtime.h>
#include <hip/hip_bf16.h>
#include <math.h>

typedef __attribute__((ext_vector_type(2))) float v2f;
typedef __attribute__((ext_vector_type(8))) float v8f;

#define NCOL 256
#define ED_K 338
#define RBF_N 16
#define SBF_L 7
#define LDSTR 258   // 256 + 2 padding floats: conflict-free b64 column reads

// ---------------------------------------------------------------------------
// Kernel 1: per-edge distance, polynomial envelope, node_rbf [E,16]
// ---------------------------------------------------------------------------
__global__ void edge_env_rbf_kernel(const float* __restrict__ pos,
                                    const int* __restrict__ eidx,
                                    float* __restrict__ ws_d,
                                    float* __restrict__ ws_env,
                                    float* __restrict__ node_rbf,
                                    int E) {
    int e = blockIdx.x * blockDim.x + threadIdx.x;
    if (e >= E) return;
    int s = eidx[e];
    int t = eidx[E + e];
    float dx = pos[3 * s + 0] - pos[3 * t + 0];
    float dy = pos[3 * s + 1] - pos[3 * t + 1];
    float dz = pos[3 * s + 2] - pos[3 * t + 2];
    float r = sqrtf(dx * dx + dy * dy + dz * dz);
    float dd = r * 0.2f;                      // r / cutoff
    float d2 = dd * dd;
    float d5 = d2 * d2 * dd;                  // dd^5
    float env = 1.0f - 21.0f * d5 + 35.0f * d5 * dd - 15.0f * d5 * d2;
    env = (dd < 1.0f) ? env : 0.0f;
    ws_d[e] = r;
    ws_env[e] = env;
    float coef = sqrtf(0.4f) * env / r;       // sqrt(2/cutoff) * env / r
    const float pc = 3.14159265358979323846f * 0.2f;  // pi / cutoff
    float* orow = node_rbf + (size_t)e * RBF_N;
#pragma unroll
    for (int n = 1; n <= RBF_N; ++n) {
        orow[n - 1] = coef * sinf((float)n * pc * r);
    }
}

// ---------------------------------------------------------------------------
// Kernel 2: FUSED 2-layer MLP on WMMA:
//   out = silu( silu((edge_attr*env) @ W_mat + b1) @ W_emb + b2 )
// One wave per 16-row tile owns the full N=256 width (16 v8f accumulators).
// The [16,256] intermediate lives in LDS between the two GEMMs, eliminating
// the 98MB+98MB HBM staging round-trip.
// Fragment layouts per CDNA5 ISA 7.12.2:
//   A (16x4 f32):   lane -> M = lane&15, K = 2*(lane>>4) + {0,1}
//   B (4x16 f32):   lane -> N = lane&15, K = 2*(lane>>4) + {0,1}
//   C/D (16x16):    VGPR v -> M = v + 8*(lane>>4), N = lane&15
// ---------------------------------------------------------------------------
__global__ __launch_bounds__(32, 1)
void fused_mlp_wmma(const float* __restrict__ A,     // edge_attr [rows, 338]
                    const float* __restrict__ B1,    // W_mat [338, 256]
                    const float* __restrict__ bias1, // [256]
                    const float* __restrict__ B2,    // W_emb [256, 256]
                    const float* __restrict__ bias2, // [256]
                    const float* __restrict__ env,   // [rows]
                    float* __restrict__ out) {       // [rows, 256]
    __shared__ float lds[16 * LDSTR];                // 16.5 KB / wave

    const int tile  = blockIdx.x;          // one wave per block (blockDim=32)
    const int lane  = threadIdx.x & 31;
    const int mloc  = lane & 15;
    const int khalf = lane >> 4;
    const long row  = (long)tile * 16 + mloc;
    const float esc = env[row];
    const float* Arow = A + row * (long)ED_K;

    v8f acc[16];
#pragma unroll
    for (int nt = 0; nt < 16; ++nt)
        acc[nt] = (v8f){0.f, 0.f, 0.f, 0.f, 0.f, 0.f, 0.f, 0.f};

    // ---------------- GEMM1: K = 338 ----------------
    constexpr int KMAIN = (ED_K / 4) * 4;  // 336
    for (int kb = 0; kb < KMAIN; kb += 4) {
        const int k0 = kb + khalf * 2;
        v2f a;
        a.x = Arow[k0] * esc;
        a.y = Arow[k0 + 1] * esc;
        const float* Bp = B1 + (long)k0 * NCOL + mloc;
#pragma unroll
        for (int nt = 0; nt < 16; ++nt) {
            v2f b;
            b.x = Bp[nt * 16];
            b.y = Bp[nt * 16 + NCOL];
            acc[nt] = __builtin_amdgcn_wmma_f32_16x16x4_f32(
                false, a, false, b, (short)0, acc[nt], false, false);
        }
    }
    {   // K tail (336..337): zero-pad beyond K=338
        const int k0 = KMAIN + khalf * 2;
        v2f a;
        a.x = (k0     < ED_K) ? Arow[k0]     * esc : 0.0f;
        a.y = (k0 + 1 < ED_K) ? Arow[k0 + 1] * esc : 0.0f;
#pragma unroll
        for (int nt = 0; nt < 16; ++nt) {
            v2f b;
            b.x = (k0     < ED_K) ? B1[(long)k0 * NCOL + nt * 16 + mloc]       : 0.0f;
            b.y = (k0 + 1 < ED_K) ? B1[(long)(k0 + 1) * NCOL + nt * 16 + mloc] : 0.0f;
            acc[nt] = __builtin_amdgcn_wmma_f32_16x16x4_f32(
                false, a, false, b, (short)0, acc[nt], false, false);
        }
    }

    // bias1 + SiLU -> LDS tile [m][k], padded row stride
#pragma unroll
    for (int nt = 0; nt < 16; ++nt) {
        const int n = nt * 16 + mloc;
        const float bv = bias1[n];
#pragma unroll
        for (int v = 0; v < 8; ++v) {
            const int m = v + khalf * 8;
            float val = acc[nt][v] + bv;
            val = val / (1.0f + expf(-val));
            lds[m * LDSTR + n] = val;
        }
        acc[nt] = (v8f){0.f, 0.f, 0.f, 0.f, 0.f, 0.f, 0.f, 0.f};
    }
    __syncthreads();   // single-wave group: forces dscnt wait before reads

    // ---------------- GEMM2: K = 256, A from LDS ----------------
    for (int kb = 0; kb < NCOL; kb += 4) {
        const int k0 = kb + khalf * 2;
        const v2f a = *(const v2f*)&lds[mloc * LDSTR + k0];  // ds_load_b64
        const float* Bp = B2 + (long)k0 * NCOL + mloc;
#pragma unroll
        for (int nt = 0; nt < 16; ++nt) {
            v2f b;
            b.x = Bp[nt * 16];
            b.y = Bp[nt * 16 + NCOL];
            acc[nt] = __builtin_amdgcn_wmma_f32_16x16x4_f32(
                false, a, false, b, (short)0, acc[nt], false, false);
        }
    }

    // bias2 + SiLU -> final output
#pragma unroll
    for (int nt = 0; nt < 16; ++nt) {
        const int n = nt * 16 + mloc;
        const float bv = bias2[n];
#pragma unroll
        for (int v = 0; v < 8; ++v) {
            const long m = (long)tile * 16 + v + khalf * 8;
            float val = acc[nt][v] + bv;
            val = val / (1.0f + expf(-val));
            out[m * NCOL + n] = val;
        }
    }
}

// ---------------------------------------------------------------------------
// Kernel 3: triplet 2D Fourier-Bessel features edge_sbf [T,112].
// One wave per triplet: scalar geometry computed redundantly per lane, then
// lanes cover the 112 outputs in contiguous 128B bursts.
// ---------------------------------------------------------------------------
__global__ void triplet_sbf_kernel(const float* __restrict__ pos,
                                   const int* __restrict__ aj,
                                   const int* __restrict__ ai,
                                   const int* __restrict__ ak,
                                   const int* __restrict__ te,
                                   const float* __restrict__ ws_d,
                                   const float* __restrict__ ws_env,
                                   float* __restrict__ out_sbf,
                                   int T) {
    int w = (blockIdx.x * blockDim.x + threadIdx.x) >> 5;
    int lane = threadIdx.x & 31;
    if (w >= T) return;
    int j = aj[w], i = ai[w], k = ak[w], e = te[w];
    float jx = pos[3 * j], jy = pos[3 * j + 1], jz = pos[3 * j + 2];
    float ix = pos[3 * i] - jx, iy = pos[3 * i + 1] - jy, iz = pos[3 * i + 2] - jz;
    float kx = pos[3 * k] - jx, ky = pos[3 * k + 1] - jy, kz = pos[3 * k + 2] - jz;
    float cosa = ix * kx + iy * ky + iz * kz;
    float cx = iy * kz - iz * ky;
    float cy = iz * kx - ix * kz;
    float cz = ix * ky - iy * kx;
    float sina = sqrtf(cx * cx + cy * cy + cz * cz);
    float theta = atan2f(sina, cosa);
    float r = ws_d[e];
    float coef = sqrtf(0.4f) * ws_env[e] / r;
    const float pc = 3.14159265358979323846f * 0.2f;
    float* orow = out_sbf + (size_t)w * (SBF_L * RBF_N);
#pragma unroll
    for (int it = 0; it < 4; ++it) {
        int idx = lane + 32 * it;
        if (idx < SBF_L * RBF_N) {
            int l  = idx >> 4;
            int rr = (idx & 15) + 1;
            orow[idx] = cosf((float)l * theta) * coef * sinf((float)rr * pc * r);
        }
    }
}

// ---------------------------------------------------------------------------
// Kernel 4: neo_edge_attr [T,128] = emb_table[x[atom_j[t]]]  (gather copy)
// One wave per row: 32 lanes x float4 = the full 512B row, fully coalesced.
// ---------------------------------------------------------------------------
__global__ void gather_emb_kernel(const float* __restrict__ emb,
                                  const int* __restrict__ x,
                                  const int* __restrict__ aj,
                                  float* __restrict__ out,
                                  int T) {
    int w = (blockIdx.x * blockDim.x + threadIdx.x) >> 5;
    int lane = threadIdx.x & 31;
    if (w >= T) return;
    int z = x[aj[w]];
    const float4* src = (const float4*)(emb + (size_t)z * 128);
    float4* dst = (float4*)(out + (size_t)w * 128);
    dst[lane] = src[lane];
}

// ---------------------------------------------------------------------------
extern "C" void kernel_launch(void* const* d_in, const int* in_sizes, int n_in,
                              void* d_out, int out_size, void* d_ws, size_t ws_size,
                              hipStream_t stream) {
    const float* atom_pos  = (const float*)d_in[0];
    const float* edge_attr = (const float*)d_in[1];
    const float* emb_table = (const float*)d_in[2];
    const float* W_mat     = (const float*)d_in[3];
    const float* b_mat     = (const float*)d_in[4];
    const float* W_emb     = (const float*)d_in[5];
    const float* b_emb     = (const float*)d_in[6];
    const int*   x         = (const int*)d_in[7];
    const int*   edge_index= (const int*)d_in[8];
    const int*   atom_j    = (const int*)d_in[9];
    const int*   atom_i    = (const int*)d_in[10];
    const int*   atom_k    = (const int*)d_in[11];
    const int*   trip_edge = (const int*)d_in[12];

    const int E = in_sizes[1] / ED_K;     // 96000
    const int T = in_sizes[9];            // 672000

    // workspace: d[E], env[E]
    float* ws_d   = (float*)d_ws;
    float* ws_env = ws_d + E;

    float* out      = (float*)d_out;
    float* out_neox = out;                                   // [E,256]
    float* out_sbf  = out_neox + (size_t)E * NCOL;           // [T,112]
    float* out_rbf  = out_sbf + (size_t)T * (SBF_L * RBF_N); // [E,16]
    float* out_nea  = out_rbf + (size_t)E * RBF_N;           // [T,128]

    // 1) edge distances / envelope / node_rbf
    edge_env_rbf_kernel<<<(E + 255) / 256, 256, 0, stream>>>(
        atom_pos, edge_index, ws_d, ws_env, out_rbf, E);

    // 2) fused 2-layer MLP (both GEMMs, intermediate tile in LDS)
    const int mtiles = E / 16;            // 6000 (E divisible by 16)
    fused_mlp_wmma<<<mtiles, 32, 0, stream>>>(
        edge_attr, W_mat, b_mat, W_emb, b_emb, ws_env, out_neox);

    // 3) triplet angular x radial basis (one wave per triplet)
    const int tblocks = (T + 7) / 8;      // 8 waves / 256-thread block
    triplet_sbf_kernel<<<tblocks, 256, 0, stream>>>(
        atom_pos, atom_j, atom_i, atom_k, trip_edge, ws_d, ws_env, out_sbf, T);

    // 4) embedding gather (one wave per 512B row)
    gather_emb_kernel<<<tblocks, 256, 0, stream>>>(
        emb_table, x, atom_j, out_nea, T);
}